// MesoLayer_31035433681485
// MI455X (gfx1250) — compile-verified
//
#include <hip/hip_runtime.h>
#include <cstdint>

// ---------------------------------------------------------------------------
// MesoLayer attention-pool, MI455X (gfx1250).
// Memory-bound: ~194MB in + ~18MB out -> ~9us floor @ 23.3TB/s; ~1.26 GFLOP
// total (0.14 TFLOP/s at the floor) => WMMA is pointless here; the CDNA5
// feature that matters is data movement:
//   primary : TDM tensor_load_to_lds — one DMA descriptor gathers the whole
//             block tile (128 batches x C dwords, global row stride 370) into
//             LDS; tracked by TENSORcnt.
//   fallback: per-lane global_load_async_to_lds_b32 (ASYNCcnt), coalesced.
// Compute: wave32 threads each own one batch element, read LDS rows.
// ---------------------------------------------------------------------------

#define GLAS  __attribute__((address_space(1)))
#define LDSAS __attribute__((address_space(3)))

typedef unsigned int v4u __attribute__((ext_vector_type(4)));
typedef int          v4i __attribute__((ext_vector_type(4)));
typedef int          v8i __attribute__((ext_vector_type(8)));

#if defined(__HIP_DEVICE_COMPILE__) && __has_builtin(__builtin_amdgcn_tensor_load_to_lds)
#define MESO_USE_TDM 1
#else
#define MESO_USE_TDM 0
#endif

__device__ __forceinline__ void async_g2l_b32(const float* g, const float* l) {
  // Low 32 bits of a generic LDS pointer are the LDS byte offset.
  const unsigned lds_off = (unsigned)(uintptr_t)l;
#if defined(__HIP_DEVICE_COMPILE__) && __has_builtin(__builtin_amdgcn_global_load_async_to_lds_b32)
  __builtin_amdgcn_global_load_async_to_lds_b32(
      (GLAS int*)(uintptr_t)g, (LDSAS int*)(uintptr_t)lds_off, 0, 0);
#else
  asm volatile("global_load_async_to_lds_b32 %0, %1, off"
               :
               : "v"(lds_off), "v"((unsigned long long)(uintptr_t)g)
               : "memory");
#endif
}

__device__ __forceinline__ void wait_async_zero() {
#if defined(__HIP_DEVICE_COMPILE__) && __has_builtin(__builtin_amdgcn_s_wait_asynccnt)
  __builtin_amdgcn_s_wait_asynccnt(0);
#else
  asm volatile("s_wait_asynccnt 0" ::: "memory");
#endif
}

__device__ __forceinline__ void wait_tensor_zero() {
#if defined(__HIP_DEVICE_COMPILE__) && __has_builtin(__builtin_amdgcn_s_wait_tensorcnt)
  __builtin_amdgcn_s_wait_tensorcnt(0);
#else
  asm volatile("s_wait_tensorcnt 0" ::: "memory");
#endif
}

#if MESO_USE_TDM
// One 2D-tile DMA: rows = nrows batch chunks of C dwords, global row stride
// 370 dwords, packed into LDS at lds_byte_off with row stride C dwords.
// D# packing per CDNA5 ISA.
__device__ __forceinline__ void tdm_load_tile(const float* gsrc, unsigned lds_byte_off,
                                              int C, int nrows) {
  const unsigned long long ga = (unsigned long long)(uintptr_t)gsrc;
  v4u g0;
  g0[0] = 1u;                                             // count=1, user load
  g0[1] = lds_byte_off;                                   // lds_addr (bytes)
  g0[2] = (unsigned)(ga & 0xFFFFFFFFull);                 // global_addr[31:0]
  g0[3] = (unsigned)((ga >> 32) & 0x01FFFFFFull)          // global_addr[56:32]
          | 0x80000000u;                                  // type=2 ("image")
  v8i g1;
  g1[0] = 0x00020000;                        // wg_mask=0, data_size=2 (4B)
  g1[1] = (C & 0xFFFF) << 16;                // tensor_dim0[15:0]
  g1[2] = ((nrows & 0xFFFF) << 16);          // tensor_dim0[31:16]=0, tensor_dim1[15:0]
  g1[3] = (C & 0xFFFF) << 16;                // tensor_dim1[31:16]=0, tile_dim0=C
  g1[4] = (nrows & 0xFFFF);                  // tile_dim1=nrows, tile_dim2=0
  g1[5] = 370;                               // tensor_dim0_stride[31:0]
  g1[6] = 0;                                 // stride0[47:32]=0, stride1 lo=0
  g1[7] = 0;
  v4i gz4 = {0, 0, 0, 0};                    // groups 2/3: 2D tile -> zero
#if __clang_major__ >= 23
  v8i gz8 = {0, 0, 0, 0, 0, 0, 0, 0};
  __builtin_amdgcn_tensor_load_to_lds(g0, g1, gz4, gz4, gz8, /*cpol*/ 0);
#else
  __builtin_amdgcn_tensor_load_to_lds(g0, g1, gz4, gz4, /*cpol*/ 0);
#endif
}
#endif

// x layout: (B, 74, 5) f32 contiguous; out layout: (B, 7, 5) f32.
// One launch per graph segment; L is compile-time so score[] stays in VGPRs.
template <int L>
__global__ __launch_bounds__(128) void meso_seg_kernel(
    const float* __restrict__ x, const float* __restrict__ w,
    float* __restrict__ out, int Bn, int segoff5 /* = seg_start*5 */,
    int outoff5 /* = seg_id*5 */) {
  constexpr int C = L * 5;  // floats per batch element in this segment
  // TDM packs rows back-to-back (stride C); async fallback uses odd stride.
  constexpr int RSTRIDE = MESO_USE_TDM ? C : (C | 1);
  extern __shared__ float smem[];

  const int t = threadIdx.x;
  const int b0 = blockIdx.x * 128;
  int nvalid = Bn - b0;
  if (nvalid > 128) nvalid = 128;

  const float* gtile = x + (size_t)b0 * 370 + (size_t)segoff5;

#if MESO_USE_TDM
  // Wave 0 issues a single Tensor-Data-Mover descriptor for the whole tile.
  if (t < 32) {
    tdm_load_tile(gtile, (unsigned)(uintptr_t)smem, C, nvalid);
    wait_tensor_zero();
  }
#else
  // Per-lane CDNA5 async copies: lane t copies dword t of every batch chunk.
  if (t < C) {
    const float* g = gtile + t;
    const float* l = smem + t;
    for (int c = 0; c < nvalid; ++c) {
      async_g2l_b32(g, l);
      g += 370;
      l += RSTRIDE;
    }
  }
#endif

  // Weight is uniform across the wave -> scalar loads, overlaps DMA traffic.
  float W[5][10];
#pragma unroll
  for (int g5 = 0; g5 < 5; ++g5)
#pragma unroll
    for (int h = 0; h < 10; ++h) W[g5][h] = w[g5 * 10 + h];

#if !MESO_USE_TDM
  wait_async_zero();  // this wave's async writes to LDS complete
#endif
  __syncthreads();    // all tile data visible to all waves

  if (t >= nvalid) return;
  const float* row = smem + t * RSTRIDE;  // this thread's (L x 5) sub-matrix

  // cs[g] = sum_j sub[j][g]
  float cs[5] = {0.f, 0.f, 0.f, 0.f, 0.f};
#pragma unroll
  for (int j = 0; j < L; ++j)
#pragma unroll
    for (int g5 = 0; g5 < 5; ++g5) cs[g5] += row[j * 5 + g5];

  // S = cs . W (1x10), then v = W . S^T (5): scores[j] = fw[j].S = sub[j].v
  float S[10];
#pragma unroll
  for (int h = 0; h < 10; ++h) {
    float a = 0.f;
#pragma unroll
    for (int g5 = 0; g5 < 5; ++g5) a += cs[g5] * W[g5][h];
    S[h] = a;
  }
  float v[5];
#pragma unroll
  for (int g5 = 0; g5 < 5; ++g5) {
    float a = 0.f;
#pragma unroll
    for (int h = 0; h < 10; ++h) a += W[g5][h] * S[h];
    v[g5] = a;
  }

  // scores + running max
  float sc[L];
  float m = -3.402823466e38f;
#pragma unroll
  for (int j = 0; j < L; ++j) {
    float a = 0.f;
#pragma unroll
    for (int g5 = 0; g5 < 5; ++g5) a += row[j * 5 + g5] * v[g5];
    sc[j] = a;
    m = fmaxf(m, a);
  }

  // softmax weights (unnormalized) + partition function
  float Z = 0.f;
#pragma unroll
  for (int j = 0; j < L; ++j) {
    const float e = __expf(sc[j] - m);
    sc[j] = e;
    Z += e;
  }
  const float rZ = 1.0f / Z;

  // pooled[g] = (sum_j e_j * sub[j][g]) / Z
  float pooled[5] = {0.f, 0.f, 0.f, 0.f, 0.f};
#pragma unroll
  for (int j = 0; j < L; ++j) {
    const float e = sc[j];
#pragma unroll
    for (int g5 = 0; g5 < 5; ++g5) pooled[g5] += e * row[j * 5 + g5];
  }

  float* o = out + (size_t)(b0 + t) * 35 + outoff5;
#pragma unroll
  for (int g5 = 0; g5 < 5; ++g5) o[g5] = pooled[g5] * rZ;
}

template <int L>
static void run_seg(const float* x, const float* w, float* out, int Bn,
                    int segoff, int segid, hipStream_t stream) {
  // Host always allocates the worst-case (odd-padded) stride so host/device
  // agree regardless of which staging path the device compile selected.
  constexpr int STRIDE_MAX = (L * 5) | 1;
  const size_t shmem = (size_t)128 * STRIDE_MAX * sizeof(float);  // <= 64,512B
  dim3 grid((unsigned)((Bn + 127) / 128)), block(128);
  meso_seg_kernel<L><<<grid, block, shmem, stream>>>(x, w, out, Bn, segoff * 5,
                                                     segid * 5);
}

extern "C" void kernel_launch(void* const* d_in, const int* in_sizes, int n_in,
                              void* d_out, int out_size, void* d_ws,
                              size_t ws_size, hipStream_t stream) {
  (void)n_in; (void)out_size; (void)d_ws; (void)ws_size;
  const float* x = (const float*)d_in[0];  // (B, 74, 5) f32
  const float* w = (const float*)d_in[1];  // (5, 10)  f32
  float* out = (float*)d_out;              // (B, 7, 5) f32
  const int Bn = in_sizes[0] / (74 * 5);

  // GRAPH_LIST = [5, 9, 9, 25, 9, 9, 12]; prefix offsets 0,5,14,23,48,57,66.
  run_seg<5>(x, w, out, Bn, 0, 0, stream);
  run_seg<9>(x, w, out, Bn, 5, 1, stream);
  run_seg<9>(x, w, out, Bn, 14, 2, stream);
  run_seg<25>(x, w, out, Bn, 23, 3, stream);
  run_seg<9>(x, w, out, Bn, 48, 4, stream);
  run_seg<9>(x, w, out, Bn, 57, 5, stream);
  run_seg<12>(x, w, out, Bn, 66, 6, stream);
}